// AttnDecoderRNN_45518063403628
// MI455X (gfx1250) — compile-verified
//
#include <hip/hip_runtime.h>
#include <hip/hip_bf16.h>

typedef __attribute__((ext_vector_type(2))) float v2f;
typedef __attribute__((ext_vector_type(8))) float v8f;

#define HD 512
#define LL 64
#define VV 32000
#define BB 32
#define TT 48
#define SOS 1

__device__ __forceinline__ float dot512(const float* __restrict__ a,
                                        const float* __restrict__ b) {
  const float4* a4 = (const float4*)a;
  const float4* b4 = (const float4*)b;
  float s = 0.f;
#pragma unroll 8
  for (int i = 0; i < HD / 4; ++i) {
    float4 x = a4[i], y = b4[i];
    s = fmaf(x.x, y.x, s);
    s = fmaf(x.y, y.y, s);
    s = fmaf(x.z, y.z, s);
    s = fmaf(x.w, y.w, s);
  }
  return s;
}

__device__ __forceinline__ float sigm(float v) { return 1.f / (1.f + __expf(-v)); }

// -------- Phase 1: recurrence --------

__global__ __launch_bounds__(256) void k0_init(const float* __restrict__ hidden,
                                               float* __restrict__ h) {
  int i = blockIdx.x * 256 + threadIdx.x;
  if (i < BB * HD) h[i] = hidden[i];
}

// blocks 0..1: attention scores [B,L]; blocks 2..49: gh = h @ W_hh.T + b_hh  [B,3H]
__global__ __launch_bounds__(256) void k1_scores_gh(
    int t, const int* __restrict__ tgt, const float* __restrict__ emb,
    const float* __restrict__ attn_W, const float* __restrict__ attn_b,
    const float* __restrict__ W_hh, const float* __restrict__ b_hh,
    const float* __restrict__ h, float* __restrict__ scores, float* __restrict__ gh) {
  if (blockIdx.x < 2) {
    for (int e = blockIdx.x * 256 + threadIdx.x; e < BB * LL; e += 512) {
      int b = e >> 6, l = e & (LL - 1);
      int tok = (t == 0) ? SOS : tgt[(t - 1) * BB + b];
      float s = attn_b[l]
              + dot512(emb + tok * HD, attn_W + l * 2 * HD)
              + dot512(h + b * HD, attn_W + l * 2 * HD + HD);
      scores[e] = s;
    }
  } else {
    int blk = blockIdx.x - 2;  // 0..47, 32 columns each
    for (int e = threadIdx.x; e < 1024; e += 256) {
      int b = e >> 5, c = blk * 32 + (e & 31);
      gh[b * (3 * HD) + c] = b_hh[c] + dot512(h + b * HD, W_hh + c * HD);
    }
  }
}

// softmax(scores) (redundant per block, tiny) then applied = aw @ encoded
__global__ __launch_bounds__(256) void k2_softmax_applied(
    const float* __restrict__ scores, const float* __restrict__ encoded,
    float* __restrict__ applied) {
  __shared__ float aw[BB * LL];
  if (threadIdx.x < BB) {
    int b = threadIdx.x;
    float m = -1e30f;
    for (int l = 0; l < LL; ++l) m = fmaxf(m, scores[b * LL + l]);
    float s = 0.f;
    for (int l = 0; l < LL; ++l) {
      float e = __expf(scores[b * LL + l] - m);
      aw[b * LL + l] = e;
      s += e;
    }
    float inv = 1.f / s;
    for (int l = 0; l < LL; ++l) aw[b * LL + l] *= inv;
  }
  __syncthreads();
  for (int e = blockIdx.x * 256 + threadIdx.x; e < BB * HD; e += 16 * 256) {
    int b = e >> 9, hh = e & (HD - 1);
    float s = 0.f;
#pragma unroll 4
    for (int l = 0; l < LL; ++l)
      s = fmaf(aw[b * LL + l], encoded[(l * BB + b) * HD + hh], s);
    applied[e] = s;
  }
}

// x = relu([embedded, applied] @ comb_W.T + comb_b)
__global__ __launch_bounds__(256) void k3_combine(
    int t, const int* __restrict__ tgt, const float* __restrict__ emb,
    const float* __restrict__ comb_W, const float* __restrict__ comb_b,
    const float* __restrict__ applied, float* __restrict__ x) {
  for (int e = blockIdx.x * 256 + threadIdx.x; e < BB * HD; e += 16 * 256) {
    int b = e >> 9, j = e & (HD - 1);
    int tok = (t == 0) ? SOS : tgt[(t - 1) * BB + b];
    float s = comb_b[j]
            + dot512(emb + tok * HD, comb_W + j * 2 * HD)
            + dot512(applied + b * HD, comb_W + j * 2 * HD + HD);
    x[e] = fmaxf(s, 0.f);
  }
}

// gx = x @ W_ih.T + b_ih ; GRU gates ; h_new -> h and Hs[t]
__global__ __launch_bounds__(256) void k4_gru(
    int t, const float* __restrict__ W_ih, const float* __restrict__ b_ih,
    const float* __restrict__ x, const float* __restrict__ gh,
    float* __restrict__ h, float* __restrict__ Hs) {
  for (int e = blockIdx.x * 256 + threadIdx.x; e < BB * HD; e += 16 * 256) {
    int b = e >> 9, j = e & (HD - 1);
    const float* xb = x + b * HD;
    float gxr = b_ih[j]          + dot512(xb, W_ih + j * HD);
    float gxz = b_ih[HD + j]     + dot512(xb, W_ih + (HD + j) * HD);
    float gxn = b_ih[2 * HD + j] + dot512(xb, W_ih + (2 * HD + j) * HD);
    const float* ghb = gh + b * 3 * HD;
    float r = sigm(gxr + ghb[j]);
    float z = sigm(gxz + ghb[HD + j]);
    float n = tanhf(gxn + r * ghb[2 * HD + j]);
    float hv = h[e];
    float hn = (1.f - z) * n + z * hv;
    h[e] = hn;
    Hs[t * BB * HD + e] = hn;
  }
}

// -------- Phase 2: logits[T*B, V] = Hs @ out_W.T + out_b (fp32 WMMA) --------
// Wave tile: 32M x 80N (2 M-tiles x 5 N-tiles). Block: 8 waves -> 32M x 640N.
// Grid: (1536/32) x (32000/640) = 48 x 50.
__global__ __launch_bounds__(256) void gemm_logits(
    const float* __restrict__ Hs, const float* __restrict__ outW,
    const float* __restrict__ out_b, float* __restrict__ out) {
  const int lane = threadIdx.x & 31;
  const int wave = threadIdx.x >> 5;
  const int l16 = lane & 15;
  const int koff = (lane >> 4) * 2;  // lanes 16..31 hold K+2,K+3 (A) / K+2,K+3 (B)
  const int mbase = blockIdx.x * 32;
  const int nbase = blockIdx.y * 640 + wave * 80;

  const float* pA0 = Hs + (mbase + l16) * HD + koff;        // M-tile 0, rows striped on lanes
  const float* pA1 = pA0 + 16 * HD;                         // M-tile 1
  const float* pB0 = outW + (nbase + l16) * HD + koff;      // B[k][n] = out_W[n][k]

  v8f acc[2][5] = {};

#pragma unroll 4
  for (int k = 0; k < HD; k += 4) {
    v2f a0 = *(const v2f*)(pA0 + k);
    v2f a1 = *(const v2f*)(pA1 + k);
#pragma unroll
    for (int j = 0; j < 5; ++j) {
      v2f bb = *(const v2f*)(pB0 + j * 16 * HD + k);
      acc[0][j] = __builtin_amdgcn_wmma_f32_16x16x4_f32(
          false, a0, false, bb, (short)0, acc[0][j], false, false);
      acc[1][j] = __builtin_amdgcn_wmma_f32_16x16x4_f32(
          false, a1, false, bb, (short)0, acc[1][j], false, false);
    }
  }

  // C/D layout: VGPR r -> row r (lanes 0-15) / row r+8 (lanes 16-31), col = lane&15
  const int rowoff = (lane >> 4) * 8;
#pragma unroll
  for (int j = 0; j < 5; ++j) {
    int col = nbase + j * 16 + l16;
    float bias = out_b[col];
#pragma unroll
    for (int mi = 0; mi < 2; ++mi) {
#pragma unroll
      for (int r = 0; r < 8; ++r) {
        int row = mbase + mi * 16 + rowoff + r;
        out[row * VV + col] = acc[mi][j][r] + bias;
      }
    }
  }
}

extern "C" void kernel_launch(void* const* d_in, const int* in_sizes, int n_in,
                              void* d_out, int out_size, void* d_ws, size_t ws_size,
                              hipStream_t stream) {
  const float* encoded = (const float*)d_in[0];
  const float* hidden  = (const float*)d_in[1];
  const int*   target  = (const int*)d_in[2];
  const float* emb     = (const float*)d_in[3];
  const float* attn_W  = (const float*)d_in[4];
  const float* attn_b  = (const float*)d_in[5];
  const float* comb_W  = (const float*)d_in[6];
  const float* comb_b  = (const float*)d_in[7];
  const float* W_ih    = (const float*)d_in[8];
  const float* W_hh    = (const float*)d_in[9];
  const float* b_ih    = (const float*)d_in[10];
  const float* b_hh    = (const float*)d_in[11];
  const float* out_W   = (const float*)d_in[12];
  const float* out_b   = (const float*)d_in[13];
  float* out = (float*)d_out;

  float* ws      = (float*)d_ws;
  float* h       = ws;            // 16384 floats
  float* scores  = ws + 16384;    // 2048
  float* applied = ws + 18432;    // 16384
  float* xbuf    = ws + 34816;    // 16384
  float* gh      = ws + 51200;    // 49152
  float* Hs      = ws + 100352;   // 786432  (total ~3.55 MB)

  k0_init<<<64, 256, 0, stream>>>(hidden, h);
  for (int t = 0; t < TT; ++t) {
    k1_scores_gh<<<50, 256, 0, stream>>>(t, target, emb, attn_W, attn_b,
                                         W_hh, b_hh, h, scores, gh);
    k2_softmax_applied<<<16, 256, 0, stream>>>(scores, encoded, applied);
    k3_combine<<<16, 256, 0, stream>>>(t, target, emb, comb_W, comb_b, applied, xbuf);
    k4_gru<<<16, 256, 0, stream>>>(t, W_ih, b_ih, xbuf, gh, h, Hs);
  }
  gemm_logits<<<dim3(48, 50), 256, 0, stream>>>(Hs, out_W, out_b, out);
}